// RandomTree_84164179132670
// MI455X (gfx1250) — compile-verified
//
#include <hip/hip_runtime.h>
#include <hip/hip_bf16.h>

// ---------------------------------------------------------------------------
// GNN-style pipeline for MI455X (gfx1250), fp32 WMMA path.
//   K1: h = features @ C                  (1M x 128) @ (128 x 64)
//   K2: gather + [x1|x2]@W -> @V -> leaky_relu -> log_softmax  (fused)
// One wave32 owns a 16-row M-tile; V_WMMA_F32_16X16X4_F32 does the math.
// ---------------------------------------------------------------------------

typedef __attribute__((ext_vector_type(2))) float v2f;
typedef __attribute__((ext_vector_type(8))) float v8f;

#define N_NODES   1000000
#define D_IN      128
#define H_DIM     64
#define NCLS      16
#define LK_ALPHA  0.2f

#define WAVES          4
#define TILE_M         16
#define ROWS_PER_BLOCK (WAVES * TILE_M)   // 64; N_NODES % 64 == 0 -> no tails

// D = A(16x4) * B(4x16) + C(16x16), fp32.
// A layout (ISA 7.12.2): lane = M + 16*(K>>1), vgpr = K&1.
// B layout (row K striped across lanes): lane = N + 16*(K>>1), vgpr = K&1.
// C/D layout: vgpr r, lanes 0-15 -> (M=r, N=lane), lanes 16-31 -> (M=r+8).
__device__ __forceinline__ v8f wmma4_f32(v2f a, v2f b, v8f c) {
    return __builtin_amdgcn_wmma_f32_16x16x4_f32(
        /*neg_a=*/false, a, /*neg_b=*/false, b,
        /*c_mod=*/(short)0, c, /*reuse_a=*/false, /*reuse_b=*/false);
}

// ---------------------------------------------------------------------------
// Kernel 1: h = features @ C
// ---------------------------------------------------------------------------
__global__ __launch_bounds__(WAVES * 32)
void k1_feat_gemm(const float* __restrict__ features,
                  const float* __restrict__ Cw,
                  float* __restrict__ h)
{
    const int lane    = threadIdx.x & 31;
    const int wave    = threadIdx.x >> 5;
    const int lane15  = lane & 15;
    const int laneHi  = lane >> 4;                       // 0 or 1
    const int baseRow = blockIdx.x * ROWS_PER_BLOCK + wave * TILE_M;

    const float* arow = features + (size_t)(baseRow + lane15) * D_IN;

    v8f acc[4] = {};                                     // 4 N-tiles of 16
    for (int kt = 0; kt < D_IN / 4; ++kt) {
        const int k = kt * 4 + 2 * laneHi;
        const float2 av = *(const float2*)(arow + k);    // (M=lane15, K=k..k+1)
        v2f a = { av.x, av.y };
        #pragma unroll
        for (int nt = 0; nt < 4; ++nt) {
            v2f b = { Cw[(size_t)k       * H_DIM + nt * 16 + lane15],
                      Cw[(size_t)(k + 1) * H_DIM + nt * 16 + lane15] };
            acc[nt] = wmma4_f32(a, b, acc[nt]);
        }
    }

    #pragma unroll
    for (int nt = 0; nt < 4; ++nt)
        #pragma unroll
        for (int r = 0; r < 8; ++r)
            h[(size_t)(baseRow + r + 8 * laneHi) * H_DIM + nt * 16 + lane15] =
                acc[nt][r];
}

// ---------------------------------------------------------------------------
// Kernel 2: gather -> [x1|x2]@W -> outs@V -> leaky_relu -> log_softmax
// ---------------------------------------------------------------------------
__global__ __launch_bounds__(WAVES * 32)
void k2_fused(const float* __restrict__ h,
              const int*   __restrict__ nbr,      // (N,2) int32
              const float* __restrict__ Ww,       // (128,64)
              const float* __restrict__ Vw,       // (64,16)
              float* __restrict__ out)            // (N,16)
{
    __shared__ float lds[WAVES][TILE_M][H_DIM + 4];      // +4 pad: conflict-free

    const int lane    = threadIdx.x & 31;
    const int wave    = threadIdx.x >> 5;
    const int lane15  = lane & 15;
    const int laneHi  = lane >> 4;
    const int baseRow = blockIdx.x * ROWS_PER_BLOCK + wave * TILE_M;

    // Per-lane neighbor pair for row M=lane15 of this tile.
    const int2 nb = ((const int2*)nbr)[baseRow + lane15];
    const float* h1 = h + (size_t)nb.x * H_DIM;
    const float* h2 = h + (size_t)nb.y * H_DIM;

    // Stage 2: outs = [x1|x2](16x128) @ W(128x64)
    v8f acc[4] = {};
    for (int kt = 0; kt < (2 * H_DIM) / 4; ++kt) {
        const int k = kt * 4 + 2 * laneHi;
        // K<64 -> x1 columns, K>=64 -> x2 columns (uniform per kt: no divergence)
        const float* src = (k < H_DIM) ? (h1 + k) : (h2 + (k - H_DIM));
        const float2 av = *(const float2*)src;
        v2f a = { av.x, av.y };
        #pragma unroll
        for (int nt = 0; nt < 4; ++nt) {
            v2f b = { Ww[(size_t)k       * H_DIM + nt * 16 + lane15],
                      Ww[(size_t)(k + 1) * H_DIM + nt * 16 + lane15] };
            acc[nt] = wmma4_f32(a, b, acc[nt]);
        }
    }

    // Re-stage outs (C/D frag) -> LDS -> A frags for the V GEMM.
    #pragma unroll
    for (int nt = 0; nt < 4; ++nt)
        #pragma unroll
        for (int r = 0; r < 8; ++r)
            lds[wave][r + 8 * laneHi][nt * 16 + lane15] = acc[nt][r];
    __syncthreads();

    // Stage 3: logits = outs(16x64) @ V(64x16)
    v8f lg = {};
    #pragma unroll
    for (int kt = 0; kt < H_DIM / 4; ++kt) {
        const int k = kt * 4 + 2 * laneHi;
        v2f a = { lds[wave][lane15][k], lds[wave][lane15][k + 1] };
        v2f b = { Vw[(size_t)k       * NCLS + lane15],
                  Vw[(size_t)(k + 1) * NCLS + lane15] };
        lg = wmma4_f32(a, b, lg);
    }

    // leaky_relu + row-wise log_softmax. Each vgpr r is one row (M = r+8*laneHi)
    // spread over a 16-lane half -> butterfly reduce with shfl_xor (masks 1..8
    // never cross the half boundary in wave32).
    #pragma unroll
    for (int r = 0; r < 8; ++r) {
        float x = lg[r];
        x = (x > 0.0f) ? x : LK_ALPHA * x;

        float mx = x;
        #pragma unroll
        for (int d = 1; d < 16; d <<= 1)
            mx = fmaxf(mx, __shfl_xor(mx, d, 32));

        float s = __expf(x - mx);
        #pragma unroll
        for (int d = 1; d < 16; d <<= 1)
            s += __shfl_xor(s, d, 32);

        out[(size_t)(baseRow + r + 8 * laneHi) * NCLS + lane15] =
            (x - mx) - __logf(s);
    }
}

// ---------------------------------------------------------------------------
extern "C" void kernel_launch(void* const* d_in, const int* in_sizes, int n_in,
                              void* d_out, int out_size, void* d_ws, size_t ws_size,
                              hipStream_t stream) {
    const float* features = (const float*)d_in[0];   // (N,128) f32
    const int*   nbr      = (const int*)  d_in[1];   // (N,2)   i32
    const float* Cw       = (const float*)d_in[2];   // (128,64) f32
    const float* Ww       = (const float*)d_in[3];   // (128,64) f32
    const float* Vw       = (const float*)d_in[4];   // (64,16)  f32
    float*       out      = (float*)d_out;           // (N,16)   f32
    float*       h        = (float*)d_ws;            // N*64*4 = 256 MB scratch

    dim3 block(WAVES * 32);
    dim3 grid(N_NODES / ROWS_PER_BLOCK);             // 15625, exact

    hipLaunchKernelGGL(k1_feat_gemm, grid, block, 0, stream, features, Cw, h);
    hipLaunchKernelGGL(k2_fused,     grid, block, 0, stream, h, nbr, Ww, Vw, out);
}